// CausalTemporalLSTM_53206054863217
// MI455X (gfx1250) — compile-verified
//
#include <hip/hip_runtime.h>
#include <hip/hip_bf16.h>

// ---------------------------------------------------------------------------
//   x : [B=256, T=1024, F=124] fp32
//   Fold linear1 into layer-0 gates: W0 = w_ih0 @ w1 : [12,124],
//     bias0 = w_ih0@b1 + b_ih0 + b_hh0.  Columns permuted to "quad" lane
//     order (lane 4k+gt <-> gate row gt*3+k) so the scan reads coalesced.
//   xp0 = x @ W0^T + bias0 : [262144,12]  -- fp32 WMMA GEMM (reads x once).
//   Scan: one wave per batch; 12 lanes = one gate each; DPP quad broadcasts
//     for the cell update, v_readlane->SGPR for the h broadcast; layer0(t),
//     layer1(t-1) and head(t-2) pipelined as independent dependency chains.
// ---------------------------------------------------------------------------

#define R_TOTAL (256 * 1024)
#define KDIM    124
#define NPAD    16
#define NG      12

typedef __attribute__((ext_vector_type(2))) float v2f;
typedef __attribute__((ext_vector_type(8))) float v8f;

// ----------------------------- fast math helpers ---------------------------

__device__ __forceinline__ float fast_rcp(float x) {
#if __has_builtin(__builtin_amdgcn_rcpf)
    return __builtin_amdgcn_rcpf(x);
#else
    return 1.0f / x;
#endif
}

// Uniform activation: bb + cc * rcp(exp(aa*x) + 1)
//   sigmoid: aa=-1, bb=0, cc= 1
//   tanh   : aa= 2, bb=1, cc=-2
__device__ __forceinline__ float actv(float x, float aa, float bb, float cc) {
    return fmaf(cc, fast_rcp(__expf(aa * x) + 1.0f), bb);
}

__device__ __forceinline__ float tanh_fast(float x) {
#if __has_builtin(__builtin_amdgcn_tanhf)
    return __builtin_amdgcn_tanhf(x);   // v_tanh_f32 on gfx1250
#else
    return 1.0f - 2.0f * fast_rcp(__expf(2.0f * x) + 1.0f);
#endif
}

// Broadcast lane Q of each quad to all 4 lanes of the quad (VALU DPP, no LDS).
template <int Q>
__device__ __forceinline__ float qbcast(float x) {
    int i = __builtin_bit_cast(int, x);
    i = __builtin_amdgcn_mov_dpp(i, Q * 0x55, 0xF, 0xF, true);  // quad_perm[Q,Q,Q,Q]
    return __builtin_bit_cast(float, i);
}

__device__ __forceinline__ float rdlane(float x, int l) {
    return __builtin_bit_cast(float, __builtin_amdgcn_readlane(__builtin_bit_cast(int, x), l));
}

// ------------------- setup: fold linear1 into layer0 gates -----------------
// W0p [124][16]; column n (n<12) holds gate row (n&3)*3 + (n>>2)  (quad order)
__global__ void __launch_bounds__(256)
setup_kernel(const float* __restrict__ w1,     // [31][124]
             const float* __restrict__ b1,     // [31]
             const float* __restrict__ w_ih0,  // [12][31]
             const float* __restrict__ b_ih0,  // [12]
             const float* __restrict__ b_hh0,  // [12]
             float* __restrict__ W0p,          // [124][16]
             float* __restrict__ bias0)        // [16]
{
    int idx = blockIdx.x * blockDim.x + threadIdx.x;
    if (idx < KDIM * NPAD) {
        int k = idx >> 4, n = idx & 15;
        float s = 0.0f;
        if (n < NG) {
            int row = (n & 3) * 3 + (n >> 2);    // quad-order permutation
            for (int d = 0; d < 31; ++d)
                s += w_ih0[row * 31 + d] * w1[d * KDIM + k];
        }
        W0p[idx] = s;
    } else if (idx < KDIM * NPAD + NPAD) {
        int n = idx - KDIM * NPAD;
        float s = 0.0f;
        if (n < NG) {
            int row = (n & 3) * 3 + (n >> 2);
            for (int d = 0; d < 31; ++d)
                s += w_ih0[row * 31 + d] * b1[d];
            s += b_ih0[row] + b_hh0[row];
        }
        bias0[n] = s;
    }
}

// ------------------------- WMMA fp32 GEMM: xp0 -----------------------------
__global__ void __launch_bounds__(256)
xp0_gemm_kernel(const float* __restrict__ x,      // [R_TOTAL][124]
                const float* __restrict__ W0p,    // [124][16]
                const float* __restrict__ bias0,  // [16]
                float* __restrict__ xp0)          // [R_TOTAL][12]
{
    const int lane   = threadIdx.x & 31;
    const int wave   = threadIdx.x >> 5;
    const int waveG  = blockIdx.x * 8 + wave;
    const int nWaves = gridDim.x * 8;
    const int m  = lane & 15;   // A row-in-tile; B/C column
    const int kh = lane >> 4;   // selects K pair {0,1} vs {2,3}

    // Preload B panel: Bk[k] = (W0^T)[4k+2*kh + {0,1}][m]   (62 VGPRs)
    v2f Bk[31];
#pragma unroll
    for (int k = 0; k < 31; ++k) {
        int kb = 4 * k + 2 * kh;
        Bk[k].x = W0p[kb * NPAD + m];
        Bk[k].y = W0p[(kb + 1) * NPAD + m];
    }
    const float bn = bias0[m];

    for (int tile = waveG; tile < R_TOTAL / 16; tile += nWaves) {
        const float* xr = x + (size_t)(tile * 16 + m) * KDIM;
        v8f acc = {0.f, 0.f, 0.f, 0.f, 0.f, 0.f, 0.f, 0.f};
#pragma unroll
        for (int k = 0; k < 31; ++k) {
            v2f a = *(const v2f*)(xr + 4 * k + 2 * kh);   // K = 4k+2kh+{0,1}
            acc = __builtin_amdgcn_wmma_f32_16x16x4_f32(
                false, a, false, Bk[k], (short)0, acc, false, false);
        }
        if (m < NG) {
#pragma unroll
            for (int v = 0; v < 8; ++v) {
                int row = tile * 16 + v + 8 * kh;         // M = v + 8*kh
                xp0[(size_t)row * NG + m] = acc[v] + bn;
            }
        }
    }
}

// ------------------- lane-parallel fused double-LSTM scan ------------------
// One wave per batch; lane 4k+gt owns gate (type gt, hidden k).
__global__ void __launch_bounds__(32)
scan_kernel(const float* __restrict__ xp0,    // [256*1024][12] (quad order)
            const float* __restrict__ w_hh0,  // [12][3]
            const float* __restrict__ w_ih1,  // [12][3]
            const float* __restrict__ w_hh1,  // [12][3]
            const float* __restrict__ b_ih1,  // [12]
            const float* __restrict__ b_hh1,  // [12]
            const float* __restrict__ w2,     // [6][3]
            const float* __restrict__ b2,     // [6]
            float* __restrict__ out)          // [256*1024][6]
{
    const int b    = blockIdx.x;          // 256 blocks of 1 wave
    const int lane = threadIdx.x & 31;
    const int gt   = lane & 3;            // gate type: 0=i 1=f 2=g 3=o
    const int kk   = lane >> 2;           // hidden index (valid for lane<12)
    const int k3   = (kk < 3) ? kk : 0;
    const int row  = gt * 3 + k3;         // PyTorch gate-row (i,f,g,o blocks)
    const int li   = (lane < NG) ? lane : 0;

    // Per-lane weights (registers)
    float w0r[3], wi1[3], wh1[3];
#pragma unroll
    for (int j = 0; j < 3; ++j) {
        w0r[j] = w_hh0[row * 3 + j];
        wi1[j] = w_ih1[row * 3 + j];
        wh1[j] = w_hh1[row * 3 + j];
    }
    const float b1v = b_ih1[row] + b_hh1[row];
    // activation constants: g-gate -> tanh, others -> sigmoid
    const float aa = (gt == 2) ?  2.0f : -1.0f;
    const float bb = (gt == 2) ?  1.0f :  0.0f;
    const float cc = (gt == 2) ? -2.0f :  1.0f;
    // head (lanes 0..5)
    const int orow = (lane < 6) ? lane : 0;
    float w2r[3] = { w2[orow * 3 + 0], w2[orow * 3 + 1], w2[orow * 3 + 2] };
    const float b2v = b2[orow];

    const float* xpb = xp0 + (size_t)b * 1024 * NG;
    float*       ob  = out + (size_t)b * 1024 * 6;

    // Broadcast state (uniform via readlane), per-quad cell state copies.
    float sh0_0 = 0.f, sh0_1 = 0.f, sh0_2 = 0.f;   // h0(t-1)
    float sh1_0 = 0.f, sh1_1 = 0.f, sh1_2 = 0.f;   // h1(t-2)
    float c0 = 0.f, c1 = 0.f;

    float xpA = xpb[li];                            // prefetch t=0

    // Iteration i: layer0 @ t=i, layer1 @ t=i-1, head @ t=i-2.
    for (int i = 0; i < 1026; ++i) {
        // ---- chain A: layer0(t=i) -------------------------------------
        float ga = fmaf(w0r[0], sh0_0, xpA);
        ga = fmaf(w0r[1], sh0_1, ga);
        ga = fmaf(w0r[2], sh0_2, ga);
        float a0 = actv(ga, aa, bb, cc);
        float i0 = qbcast<0>(a0), f0 = qbcast<1>(a0);
        float g0 = qbcast<2>(a0), o0 = qbcast<3>(a0);
        c0 = fmaf(f0, c0, i0 * g0);
        float h0n = o0 * tanh_fast(c0);

        // ---- chain B: layer1(t=i-1)  (reads OLD sh0 = h0(i-1), sh1 = h1(i-2))
        float gb = fmaf(wi1[0], sh0_0, b1v);
        gb = fmaf(wi1[1], sh0_1, gb);
        gb = fmaf(wi1[2], sh0_2, gb);
        gb = fmaf(wh1[0], sh1_0, gb);
        gb = fmaf(wh1[1], sh1_1, gb);
        gb = fmaf(wh1[2], sh1_2, gb);
        float a1 = actv(gb, aa, bb, cc);
        float i1 = qbcast<0>(a1), f1 = qbcast<1>(a1);
        float g1 = qbcast<2>(a1), o1 = qbcast<3>(a1);
        c1 = fmaf(f1, c1, i1 * g1);
        float h1n = o1 * tanh_fast(c1);

        // ---- chain C: head(t=i-2)  (reads sh1 = h1(i-2)) ---------------
        float ho = fmaf(w2r[0], sh1_0, b2v);
        ho = fmaf(w2r[1], sh1_1, ho);
        ho = fmaf(w2r[2], sh1_2, ho);
        float outv = actv(ho, -1.0f, 0.0f, 1.0f);   // sigmoid
        if ((i >= 2) && (lane < 6))
            ob[(size_t)(i - 2) * 6 + lane] = outv;

        // ---- prefetch next xp row --------------------------------------
        int tn = (i + 1 < 1024) ? (i + 1) : 1023;
        float xpN = xpb[(size_t)tn * NG + li];

        // ---- rotate broadcast state ------------------------------------
        sh0_0 = rdlane(h0n, 0); sh0_1 = rdlane(h0n, 4); sh0_2 = rdlane(h0n, 8);
        sh1_0 = rdlane(h1n, 0); sh1_1 = rdlane(h1n, 4); sh1_2 = rdlane(h1n, 8);
        xpA = xpN;
    }
}

// ---------------------------------------------------------------------------

extern "C" void kernel_launch(void* const* d_in, const int* in_sizes, int n_in,
                              void* d_out, int out_size, void* d_ws, size_t ws_size,
                              hipStream_t stream) {
    const float* x     = (const float*)d_in[0];
    const float* w1    = (const float*)d_in[1];
    const float* b1    = (const float*)d_in[2];
    const float* w_ih0 = (const float*)d_in[3];
    const float* w_hh0 = (const float*)d_in[4];
    const float* b_ih0 = (const float*)d_in[5];
    const float* b_hh0 = (const float*)d_in[6];
    const float* w_ih1 = (const float*)d_in[7];
    const float* w_hh1 = (const float*)d_in[8];
    const float* b_ih1 = (const float*)d_in[9];
    const float* b_hh1 = (const float*)d_in[10];
    const float* w2    = (const float*)d_in[11];
    const float* b2    = (const float*)d_in[12];
    float* out = (float*)d_out;

    char*  ws    = (char*)d_ws;
    float* W0p   = (float*)ws;                       // 124*16 floats
    float* bias0 = (float*)(ws + KDIM * NPAD * 4);   // 16 floats
    float* xp0   = (float*)(ws + 8192);              // 262144*12 floats (~12.6 MB)

    setup_kernel<<<8, 256, 0, stream>>>(w1, b1, w_ih0, b_ih0, b_hh0, W0p, bias0);
    xp0_gemm_kernel<<<256, 256, 0, stream>>>(x, W0p, bias0, xp0);
    scan_kernel<<<256, 32, 0, stream>>>(xp0, w_hh0, w_ih1, w_hh1,
                                        b_ih1, b_hh1, w2, b2, out);
}